// GCN_8727373545893
// MI455X (gfx1250) — compile-verified
//
#include <hip/hip_runtime.h>

typedef __attribute__((ext_vector_type(2))) float v2f;
typedef __attribute__((ext_vector_type(8))) float v8f;

// ---------------- utility kernels ----------------

__global__ void zero_f32_kernel(float* __restrict__ p, long n) {
  long i = (long)blockIdx.x * blockDim.x + threadIdx.x;
  long stride = (long)gridDim.x * blockDim.x;
  for (; i < n; i += stride) p[i] = 0.0f;
}

__global__ void degree_kernel(const int* __restrict__ src, const int* __restrict__ dst,
                              float* __restrict__ deg_out, float* __restrict__ deg_in, int E) {
  int e = blockIdx.x * blockDim.x + threadIdx.x;
  if (e >= E) return;
  atomicAdd(deg_out + src[e], 1.0f);
  atomicAdd(deg_in + dst[e], 1.0f);
}

__global__ void norm_kernel(float* __restrict__ a, float* __restrict__ b, int n) {
  int i = blockIdx.x * blockDim.x + threadIdx.x;
  if (i >= n) return;
  float d = a[i];
  a[i] = (d > 0.0f) ? rsqrtf(d) : 1.0f;
  d = b[i];
  b[i] = (d > 0.0f) ? rsqrtf(d) : 1.0f;
}

// ---------------- WMMA fp32 GEMM:  D[M x NC] = (A[M x K] * scale[M]) @ W[K x NC] ----------------
// W is staged into LDS in a *pair-swizzled* layout: element W[kk][n] lives at LDS index
// ((kk>>1)*NC + n)*2 + (kk&1), so the WMMA B fragment {W[kk][n], W[kk+1][n]} (kk even) is one
// aligned 8-byte ds_load_b64 straight into an even VGPR pair — no shuffle movs. The per-lane
// LDS base is loop-invariant; k/t offsets are compile-time immediates (max 64384 < 64Ki).
//
// Each wave owns a 16-row strip and computes all NC/16 column tiles: the A fragment
// (one global_load_b64 + one v_pk_mul_f32 for the degree norm) feeds NT WMMAs.
//
// V_WMMA_F32_16X16X4_F32 lane layouts (ISA 7.12.2):
//   A 16x4 : lanes 0-15 -> M=0..15 {K=k,k+1}; lanes 16-31 -> M=0..15 {K=k+2,k+3}
//   B 4x16 : lanes 0-15 -> N=0..15 rows {K=k,k+1}; lanes 16-31 -> N=0..15 rows {K=k+2,k+3}
//   C/D    : VGPR r -> lanes 0-15 (M=r,N=lane); lanes 16-31 (M=8+r,N=lane-16)
template <int K, int NC>
__global__ __launch_bounds__(256) void wmma_gemm_scaled(const float* __restrict__ A,
                                                        const float* __restrict__ scale,
                                                        const float* __restrict__ W,
                                                        float* __restrict__ D, int M) {
  constexpr int NT = NC / 16;  // column tiles per wave
  __shared__ __align__(16) float sW[K * NC];

  // cooperative stage of W into LDS, pair-swizzled:
  // thread handles (kh, n..n+1): reads float2 from rows 2kh and 2kh+1, writes interleaved float4.
  {
    constexpr int tot = (K / 2) * (NC / 2);
    for (int idx = threadIdx.x; idx < tot; idx += 256) {
      int kh = idx / (NC / 2);
      int n = (idx - kh * (NC / 2)) * 2;
      float2 r0 = *(const float2*)(W + (2 * kh) * NC + n);
      float2 r1 = *(const float2*)(W + (2 * kh + 1) * NC + n);
      float4 v;
      v.x = r0.x; v.y = r1.x;  // (kk=2kh,   n)   , (kk=2kh+1, n)
      v.z = r0.y; v.w = r1.y;  // (kk=2kh,   n+1) , (kk=2kh+1, n+1)
      *(float4*)(sW + (kh * NC + n) * 2) = v;
    }
  }
  __syncthreads();

  const int lane = threadIdx.x & 31;
  const int wave = threadIdx.x >> 5;
  const int m0 = (blockIdx.x * 8 + wave) << 4;
  if (m0 >= M) return;  // wave-uniform: EXEC stays all-1s for WMMA

  const int lm = lane & 15;
  const int koff = (lane >> 4) << 1;  // 0 for lanes 0-15, 2 for lanes 16-31
  const int row = m0 + lm;

  const float s = scale[row];
  const float* __restrict__ arow = A + (long)row * K;
  // loop-invariant LDS base: fragment for (k, t) sits at + (k*NC + t*32) elements
  const float* __restrict__ wbase = sW + ((koff >> 1) * NC + lm) * 2;

  v8f c[NT];
#pragma unroll
  for (int t = 0; t < NT; ++t) c[t] = (v8f){};

#pragma unroll 4
  for (int k = 0; k < K; k += 4) {
    v2f a = *(const v2f*)(arow + k + koff);  // 8B aligned (k+koff even)
    a.x *= s;
    a.y *= s;
#pragma unroll
    for (int t = 0; t < NT; ++t) {
      v2f b = *(const v2f*)(wbase + k * NC + t * 32);  // ds_load_b64, imm offset
      // 8 args: (neg_a, A, neg_b, B, c_mod, C, reuse_a, reuse_b)
      c[t] = __builtin_amdgcn_wmma_f32_16x16x4_f32(false, a, false, b, (short)0,
                                                   c[t], false, false);
    }
  }

  float* __restrict__ drow = D + (long)(m0 + ((lane >> 4) << 3)) * NC + lm;
#pragma unroll
  for (int t = 0; t < NT; ++t) {
#pragma unroll
    for (int r = 0; r < 8; ++r) drow[r * NC + t * 16] = c[t][r];
  }
}

// ---------------- wave-per-edge scatter-add:  agg[dst[e], :] += h[src[e], :] ----------------
template <int F>
__global__ void scatter_add_kernel(const float* __restrict__ h,
                                   const int* __restrict__ src,
                                   const int* __restrict__ dst,
                                   float* __restrict__ agg, int E) {
  const int lane = threadIdx.x & 31;
  const int e = blockIdx.x * (blockDim.x >> 5) + (threadIdx.x >> 5);
  if (e >= E) return;
  const float* __restrict__ hrow = h + (long)src[e] * F;
  float* __restrict__ arow = agg + (long)dst[e] * F;
#pragma unroll
  for (int j = 0; j < F / 32; ++j) {
    int f = lane + 32 * j;
    atomicAdd(arow + f, hrow[f]);
  }
}

// ---------------- finalize: out = [relu](agg * in_norm[i] + bias[j]), float4 ----------------
template <int F, bool RELU>
__global__ void finalize_kernel(const float* __restrict__ agg,
                                const float* __restrict__ norm_in,
                                const float* __restrict__ bias,
                                float* __restrict__ out, long total4) {
  constexpr int SHIFT = (F == 128) ? 7 : 6;
  long idx4 = (long)blockIdx.x * blockDim.x + threadIdx.x;
  if (idx4 >= total4) return;
  long base = idx4 << 2;
  long i = base >> SHIFT;
  int j = (int)(base & (F - 1));
  float nv = norm_in[i];
  float4 av = *(const float4*)(agg + base);
  float4 bv = *(const float4*)(bias + j);
  float4 r;
  r.x = av.x * nv + bv.x;
  r.y = av.y * nv + bv.y;
  r.z = av.z * nv + bv.z;
  r.w = av.w * nv + bv.w;
  if (RELU) {
    r.x = fmaxf(r.x, 0.0f);
    r.y = fmaxf(r.y, 0.0f);
    r.z = fmaxf(r.z, 0.0f);
    r.w = fmaxf(r.w, 0.0f);
  }
  *(float4*)(out + base) = r;
}

// ---------------- host launcher ----------------
extern "C" void kernel_launch(void* const* d_in, const int* in_sizes, int n_in,
                              void* d_out, int out_size, void* d_ws, size_t ws_size,
                              hipStream_t stream) {
  const float* in_feat = (const float*)d_in[0];
  const int*   src     = (const int*)d_in[1];
  const int*   dst     = (const int*)d_in[2];
  const float* W1      = (const float*)d_in[3];
  const float* b1      = (const float*)d_in[4];
  const float* W2      = (const float*)d_in[5];
  const float* b2      = (const float*)d_in[6];

  const int IN = 128, H = 128, OUT = 64;
  const int N = in_sizes[0] / IN;  // 100000
  const int E = in_sizes[1];       // 1600000

  // workspace layout (base 256B aligned; 2*N*4 = 800000 is 256-aligned)
  char* ws = (char*)d_ws;
  float* norm_out = (float*)ws;                                    // N
  float* norm_in  = norm_out + N;                                  // N
  float* hbuf     = (float*)(ws + (size_t)2 * N * sizeof(float));  // N*128
  float* agg      = hbuf + (long)N * H;                            // N*128

  float* out_h2 = (float*)d_out;           // N*64  (tuple elem 0)
  float* out_z  = out_h2 + (long)N * OUT;  // N*128 (tuple elem 1)

  // 1) degrees -> norms
  zero_f32_kernel<<<2048, 256, 0, stream>>>(norm_out, 2L * N);
  degree_kernel<<<(E + 255) / 256, 256, 0, stream>>>(src, dst, norm_out, norm_in, E);
  norm_kernel<<<(N + 255) / 256, 256, 0, stream>>>(norm_out, norm_in, N);

  const int mblocks = (N / 16 + 7) / 8;  // 8 waves (16-row strips) per block

  // 2) layer 1: h1 = (x * out_norm) @ W1 ; aggregate ; z = relu(...)
  wmma_gemm_scaled<128, 128><<<mblocks, 256, 0, stream>>>(in_feat, norm_out, W1, hbuf, N);
  zero_f32_kernel<<<4096, 256, 0, stream>>>(agg, (long)N * H);
  scatter_add_kernel<128><<<(E + 7) / 8, 256, 0, stream>>>(hbuf, src, dst, agg, E);
  finalize_kernel<128, true><<<(int)(((long)N * H / 4 + 255) / 256), 256, 0, stream>>>(
      agg, norm_in, b1, out_z, (long)N * H / 4);

  // 3) layer 2: h2pre = (z * out_norm) @ W2 ; aggregate ; h2 = ...
  wmma_gemm_scaled<128, 64><<<mblocks, 256, 0, stream>>>(out_z, norm_out, W2, hbuf, N);
  zero_f32_kernel<<<4096, 256, 0, stream>>>(agg, (long)N * OUT);
  scatter_add_kernel<64><<<(E + 7) / 8, 256, 0, stream>>>(hbuf, src, dst, agg, E);
  finalize_kernel<64, false><<<(int)(((long)N * OUT / 4 + 255) / 256), 256, 0, stream>>>(
      agg, norm_in, b2, out_h2, (long)N * OUT / 4);
}